// MultiHeadAttentionQuantum_65481071395483
// MI455X (gfx1250) — compile-verified
//
#include <hip/hip_runtime.h>
#include <hip/hip_bf16.h>
#include <math.h>

// Fused quantum multi-head attention for MI455X (gfx1250, wave32).
//
// Closed-form reduction of the 8-wire circuit:
//   <Z_w> (w>=1) = prod_{u=0..w} cos(t_u + theta_u)   (prefix product)
//   <Z_0>        = prod_{u=1..7} cos(t_u + theta_u)
// All four 8192x64x64 GEMMs run on V_WMMA_F32_16X16X4_F32 (f32 keeps the
// precision needed through the cos/prefix-product nonlinearity).
// One wave handles one 16-token tile end-to-end; 16 consecutive tokens
// produce 16 *complete* rows (m = h*64 + s>>3) of the scrambled out2 tensor,
// so the final out2 @ Wo^T GEMM fuses into the same wave.

typedef __attribute__((ext_vector_type(2))) float v2f;
typedef __attribute__((ext_vector_type(8))) float v8f;

#define NW 8
#define E_DIM 64
#define S_DIM 512
#define TOK_TILE 16

__device__ __forceinline__ v8f wmma_f32_k4(v2f a, v2f b, v8f c) {
  // 8 args: (neg_a, A, neg_b, B, c_mod, C, reuse_a, reuse_b)
  return __builtin_amdgcn_wmma_f32_16x16x4_f32(false, a, false, b, (short)0, c,
                                               false, false);
}

__global__ __launch_bounds__(32) void qattn_fused_kernel(
    const float* __restrict__ x, const float* __restrict__ Wq,
    const float* __restrict__ Wk, const float* __restrict__ Wv,
    const float* __restrict__ Wo, const float* __restrict__ theta,
    float* __restrict__ y) {
  // t_ld rows padded to 68 floats (272B = 16B aligned, spreads LDS banks)
  __shared__ float t_ld[3][TOK_TILE][E_DIM + 4];  // tq/tk/tv tiles
  __shared__ float qkv[3][TOK_TILE][NW][NW];      // [mat][token][head][wire]
  __shared__ float o2[TOK_TILE][E_DIM + 8];       // scrambled out2 rows (72: 8B-aligned rows)

  const int L = threadIdx.x;  // 0..31
  const int lo = L & 15;
  const int hi = L >> 4;
  const int T = blockIdx.x;          // 512 tiles total
  const int b = T >> 5;              // 32 tiles per batch element
  const int s0 = (T & 31) * TOK_TILE;
  const int sg0 = (T & 31) * 2;      // s>>3 group base (2 groups per tile)

  const float* xt = x + (size_t)(b * S_DIM + s0) * E_DIM;
  const float* Ws[3] = {Wq, Wk, Wv};

  // ---------- Step 1: t = x_tile @ W^T  (V_WMMA_F32_16X16X4_F32) ----------
  // A frag: lane holds x[m0+lo][k0+2*hi .. +1]; B frag: B[k][n]=W[n][k] ->
  // lane holds W[n0+lo][k0+2*hi .. +1]. Same addressing for both.
#pragma unroll
  for (int mtx = 0; mtx < 3; ++mtx) {
    const float* W = Ws[mtx];
#pragma unroll
    for (int n0 = 0; n0 < E_DIM; n0 += 16) {
      v8f acc = {0.f, 0.f, 0.f, 0.f, 0.f, 0.f, 0.f, 0.f};
#pragma unroll
      for (int k0 = 0; k0 < E_DIM; k0 += 4) {
        v2f a = *(const v2f*)(xt + lo * E_DIM + k0 + 2 * hi);
        v2f bb = *(const v2f*)(W + (n0 + lo) * E_DIM + k0 + 2 * hi);
        acc = wmma_f32_k4(a, bb, acc);
      }
      // D layout: VGPR r -> row r (lanes 0-15) / row r+8 (lanes 16-31), col=lane&15
#pragma unroll
      for (int r = 0; r < 8; ++r) t_ld[mtx][r + 8 * hi][n0 + lo] = acc[r];
    }
  }

  float th[NW];
#pragma unroll
  for (int w = 0; w < NW; ++w) th[w] = theta[w];

  // ---------- Step 2: quantum projection (closed form) ----------
  // 16 tokens x 8 heads = 128 pairs, 4 per lane.
#pragma unroll
  for (int i = 0; i < 4; ++i) {
    int p = L + 32 * i;
    int tok = p >> 3;
    int h = p & 7;
#pragma unroll
    for (int mtx = 0; mtx < 3; ++mtx) {
      float c[NW];
#pragma unroll
      for (int w = 0; w < NW; ++w)
        c[w] = __cosf(t_ld[mtx][tok][h * 8 + w] + th[w]);
      float z[NW];
      float pr = c[0];
#pragma unroll
      for (int w = 1; w < NW; ++w) {
        pr *= c[w];
        z[w] = pr;  // <Z_w> = prod_{u<=w} c_u
      }
      float pt = 1.f;
#pragma unroll
      for (int w = 1; w < NW; ++w) pt *= c[w];
      z[0] = pt;  // <Z_0> = prod_{u=1..7} c_u
#pragma unroll
      for (int w = 0; w < NW; ++w) qkv[mtx][tok][h][w] = z[w];
    }
  }

  // ---------- Step 3: per-token 8x8 attention (mixes HEADS, d_k=8) ----------
  // 128 (token, query-head) rows, 4 per lane. Row-wise softmax.
#pragma unroll
  for (int i = 0; i < 4; ++i) {
    int p = L + 32 * i;
    int tok = p >> 3;
    int ih = p & 7;
    float qr[NW];
#pragma unroll
    for (int w = 0; w < NW; ++w) qr[w] = qkv[0][tok][ih][w];
    float sc[NW];
    float mx = -3.0e38f;
#pragma unroll
    for (int j = 0; j < NW; ++j) {
      float d = 0.f;
#pragma unroll
      for (int w = 0; w < NW; ++w) d += qr[w] * qkv[1][tok][j][w];
      d *= 0.35355339059327373f;  // 1/sqrt(8)
      sc[j] = d;
      mx = fmaxf(mx, d);
    }
    float se = 0.f;
#pragma unroll
    for (int j = 0; j < NW; ++j) {
      sc[j] = __expf(sc[j] - mx);
      se += sc[j];
    }
    float inv = 1.f / se;
    float orow[NW];
#pragma unroll
    for (int w = 0; w < NW; ++w) orow[w] = 0.f;
#pragma unroll
    for (int j = 0; j < NW; ++j) {
      float a = sc[j] * inv;
#pragma unroll
      for (int w = 0; w < NW; ++w) orow[w] += a * qkv[2][tok][j][w];
    }
    // Scrambled layout: row r2 = head*2 + (tok>>3), col = (tok&7)*8 + w
    int r2 = ih * 2 + (tok >> 3);
    int colb = (tok & 7) * 8;
#pragma unroll
    for (int w = 0; w < NW; ++w) o2[r2][colb + w] = orow[w];
  }

  // ---------- Step 4: y rows = o2 @ Wo^T (WMMA), scatter-store ----------
#pragma unroll
  for (int n0 = 0; n0 < E_DIM; n0 += 16) {
    v8f acc = {0.f, 0.f, 0.f, 0.f, 0.f, 0.f, 0.f, 0.f};
#pragma unroll
    for (int k0 = 0; k0 < E_DIM; k0 += 4) {
      v2f a = *(const v2f*)&o2[lo][k0 + 2 * hi];
      v2f bb = *(const v2f*)(Wo + (n0 + lo) * E_DIM + k0 + 2 * hi);
      acc = wmma_f32_k4(a, bb, acc);
    }
    int n = n0 + lo;
#pragma unroll
    for (int r = 0; r < 8; ++r) {
      int rr = r + 8 * hi;                       // local out2 row
      int m = (rr >> 1) * 64 + sg0 + (rr & 1);   // global row: h*64 + s>>3
      y[(size_t)b * (S_DIM * E_DIM) + (size_t)m * E_DIM + n] = acc[r];
    }
  }
}

extern "C" void kernel_launch(void* const* d_in, const int* in_sizes, int n_in,
                              void* d_out, int out_size, void* d_ws,
                              size_t ws_size, hipStream_t stream) {
  (void)in_sizes; (void)n_in; (void)out_size; (void)d_ws; (void)ws_size;
  const float* x = (const float*)d_in[0];
  const float* Wq = (const float*)d_in[1];
  const float* Wk = (const float*)d_in[2];
  const float* Wv = (const float*)d_in[3];
  const float* Wo = (const float*)d_in[4];
  const float* theta = (const float*)d_in[5];
  float* y = (float*)d_out;
  dim3 grid(512), block(32);  // one wave32 per 16-token tile
  hipLaunchKernelGGL(qattn_fused_kernel, grid, block, 0, stream, x, Wq, Wk, Wv,
                     Wo, theta, y);
}